// MambaBlock_12395275616471
// MI455X (gfx1250) — compile-verified
//
#include <hip/hip_runtime.h>
#include <hip/hip_bf16.h>

// ---------------------------------------------------------------------------
// Mamba block forward for MI455X (gfx1250), fp32 end-to-end.
// GEMMs on the fp32 WMMA path (V_WMMA_F32_16X16X4_F32).
// Each wave computes a 32x(16*JSUB) output tile: 2 M-tiles x JSUB N-tiles,
// fully specialized at compile time so the K-loop is branch-free and the
// accumulators never spill.  Tile indices are scalarized via readfirstlane
// so addressing/epilogue stay on the SALU path.
// ---------------------------------------------------------------------------

#define LSEQ   1024
#define DMODEL 2048
#define DINNER 4096
#define DSTATE 16
#define DCONV  4
#define DTRANK 128
#define XPDIM  (DTRANK + 2 * DSTATE)   // 160

typedef __attribute__((ext_vector_type(2))) float v2f;
typedef __attribute__((ext_vector_type(8))) float v8f;

__device__ __forceinline__ v8f wmma_f32_16x16x4(v2f a, v2f b, v8f c) {
    // 8 args: (neg_a, A, neg_b, B, c_mod, C, reuse_a, reuse_b)
    return __builtin_amdgcn_wmma_f32_16x16x4_f32(
        false, a, false, b, (short)0, c, false, false);
}

__device__ __forceinline__ float silu_f(float v) {
    return v / (1.0f + expf(-v));
}

// ---------------------------------------------------------------------------
// fp32 WMMA GEMM:  C[M,N] = A[M,K] * W[N,K]^T  (+bias, +softplus)
//   KDIM  : compile-time K (multiple of 4)
//   JSUB  : 16-col subtiles per wave (N must equal strips * 16 * JSUB)
//   MROWS : 16-row tiles per wave    (M must equal mtiles * 16 * MROWS)
//   SPLIT : cols [0,N0w) -> C0 (ld N0w), cols [N0w,N) -> C1 (ld N1w)
// Launch exactly (M/(16*MROWS)) * (N/(16*JSUB)) waves, 256-thread blocks.
// ---------------------------------------------------------------------------
template<int KDIM, int JSUB, int MROWS, bool BIAS, bool SOFTPLUS, bool SPLIT>
__global__ void gemm_wmma_f32(const float* __restrict__ A, int lda,
                              const float* __restrict__ W,
                              float* __restrict__ C0, int N0w,
                              float* __restrict__ C1, int N1w,
                              const float* __restrict__ bias,
                              int tilesN)
{
    const int lane = threadIdx.x & 31;
    // Wave-invariant by construction; readfirstlane makes it provably scalar
    // so tile math / epilogue selects stay on the SALU path.
    const int wid  = __builtin_amdgcn_readfirstlane(
                         (blockIdx.x * blockDim.x + threadIdx.x) >> 5);
    const int tm   = wid / tilesN;             // scalar
    const int tn   = wid % tilesN;             // scalar

    const int l15   = lane & 15;
    const int khalf = (lane >> 4) << 1;        // 0 or 2: this lane's K sub-pair
    const int base_m = tm * (16 * MROWS);      // scalar
    const int base_n = tn * (16 * JSUB);       // scalar

    // Fragment base pointers (per ISA layout):
    //   A 16x4: lane = (K/2)*16 + M, 2 VGPRs over K%2
    //   B 4x16: lane = (K/2)*16 + N, 2 VGPRs over K%2; B[k][n] = W[n*K + k]
    const float* Ap[MROWS];
#pragma unroll
    for (int mt = 0; mt < MROWS; ++mt)
        Ap[mt] = A + (size_t)(base_m + mt * 16 + l15) * lda + khalf;

    const float* Wp[JSUB];
#pragma unroll
    for (int j = 0; j < JSUB; ++j)
        Wp[j] = W + (size_t)(base_n + j * 16 + l15) * KDIM + khalf;

    v8f acc[MROWS][JSUB] = {};

#pragma unroll 1
    for (int k = 0; k < KDIM; k += 4) {
        v2f a[MROWS];
#pragma unroll
        for (int mt = 0; mt < MROWS; ++mt)
            a[mt] = *(const v2f*)(Ap[mt] + k);
#pragma unroll
        for (int j = 0; j < JSUB; ++j) {
            const v2f b = *(const v2f*)(Wp[j] + k);
#pragma unroll
            for (int mt = 0; mt < MROWS; ++mt)
                acc[mt][j] = wmma_f32_16x16x4(a[mt], b, acc[mt][j]);
        }
    }

    // C/D layout: VGPR v holds row M = (lane>>4)*8 + v, column N = lane&15
#pragma unroll
    for (int mt = 0; mt < MROWS; ++mt) {
        const int r0 = base_m + mt * 16 + (lane >> 4) * 8;
#pragma unroll
        for (int j = 0; j < JSUB; ++j) {
            const int nb   = base_n + j * 16;  // scalar
            const int ncol = nb + l15;

            float* Cp;
            int col, ldc;
            if (SPLIT && nb >= N0w) {          // scalar select / branch
                Cp = C1; col = ncol - N0w; ldc = N1w;
            } else {
                Cp = C0; col = ncol;       ldc = N0w;
            }
#pragma unroll
            for (int v = 0; v < 8; ++v) {
                float val = acc[mt][j][v];
                if (BIAS)     val += bias[ncol];
                if (SOFTPLUS) val = (val > 20.0f) ? val : log1pf(expf(val));
                Cp[(size_t)(r0 + v) * ldc + col] = val;
            }
        }
    }
}

// ---------------------------------------------------------------------------
// Depthwise causal conv1d (width 4) + bias + silu.
//   x[l,d] = silu( sum_t w[d,t] * x_pre[l-3+t, d] + b[d] )
// ---------------------------------------------------------------------------
__global__ void conv_silu_kernel(const float* __restrict__ xpre,
                                 const float* __restrict__ w,
                                 const float* __restrict__ b,
                                 float* __restrict__ xout)
{
    const int idx = blockIdx.x * blockDim.x + threadIdx.x;
    const int d = idx & (DINNER - 1);
    const int l = idx >> 12;                  // DINNER == 4096 == 2^12
    float acc = b[d];
#pragma unroll
    for (int t = 0; t < DCONV; ++t) {
        const int ls = l - (DCONV - 1) + t;
        if (ls >= 0) acc += w[d * DCONV + t] * xpre[(size_t)ls * DINNER + d];
    }
    xout[idx] = silu_f(acc);
}

// ---------------------------------------------------------------------------
// Selective scan. One thread per (channel d, state n); the 16 states live in
// 16 lanes of a wave32 half.  y = sum_n h*C via __shfl_xor tree; lane n==0
// fuses (+ x*D) * silu(z) -> yg.
// ---------------------------------------------------------------------------
__global__ void scan_kernel(const float* __restrict__ dt,
                            const float* __restrict__ xp,
                            const float* __restrict__ x,
                            const float* __restrict__ z,
                            const float* __restrict__ A_log,
                            const float* __restrict__ Dp,
                            float* __restrict__ yg)
{
    const int t = blockIdx.x * blockDim.x + threadIdx.x;
    const int d = t >> 4;
    const int n = t & 15;

    const float Aval = -expf(A_log[d * DSTATE + n]);
    const float Dv   = Dp[d];
    float h = 0.0f;

    for (int l = 0; l < LSEQ; ++l) {
        const float dtv = dt[(size_t)l * DINNER + d];
        const float Bv  = xp[(size_t)l * XPDIM + DTRANK + n];
        const float Cv  = xp[(size_t)l * XPDIM + DTRANK + DSTATE + n];
        const float xv  = x[(size_t)l * DINNER + d];

        const float dA = expf(dtv * Aval);
        h = dA * h + dtv * Bv * xv;

        float p = h * Cv;                     // reduce over 16 states
        p += __shfl_xor(p, 1, 16);
        p += __shfl_xor(p, 2, 16);
        p += __shfl_xor(p, 4, 16);
        p += __shfl_xor(p, 8, 16);

        if (n == 0) {
            const float zv = z[(size_t)l * DINNER + d];
            yg[(size_t)l * DINNER + d] = (p + xv * Dv) * silu_f(zv);
        }
    }
}

// ---------------------------------------------------------------------------
// Host-side launch.  Inputs (setup_inputs order):
//  0:index 1:hidden 2:in_proj_w 3:conv_w 4:conv_b 5:x_proj_w
//  6:dt_proj_w 7:dt_proj_b 8:A_log 9:D 10:out_proj_w
// d_out = [ out (1024x2048) | x_pre (1024x4096) ]  (tuple return order)
// ---------------------------------------------------------------------------
extern "C" void kernel_launch(void* const* d_in, const int* in_sizes, int n_in,
                              void* d_out, int out_size, void* d_ws, size_t ws_size,
                              hipStream_t stream)
{
    const float* hidden  = (const float*)d_in[1];
    const float* in_w    = (const float*)d_in[2];
    const float* conv_w  = (const float*)d_in[3];
    const float* conv_b  = (const float*)d_in[4];
    const float* xproj_w = (const float*)d_in[5];
    const float* dt_w    = (const float*)d_in[6];
    const float* dt_b    = (const float*)d_in[7];
    const float* A_log   = (const float*)d_in[8];
    const float* Dp      = (const float*)d_in[9];
    const float* out_w   = (const float*)d_in[10];

    float* out  = (float*)d_out;                        // [1024, 2048]
    float* xpre = out + (size_t)LSEQ * DMODEL;          // [1024, 4096]

    float* ws    = (float*)d_ws;
    float* ws_z  = ws;                                  // [1024, 4096]
    float* ws_x  = ws_z  + (size_t)LSEQ * DINNER;       // [1024, 4096]
    float* ws_xp = ws_x  + (size_t)LSEQ * DINNER;       // [1024, 160]
    float* ws_dt = ws_xp + (size_t)LSEQ * XPDIM;        // [1024, 4096]
    float* ws_yg = ws_dt + (size_t)LSEQ * DINNER;       // [1024, 4096]

    const int BLK = 256;                                // 8 waves / block

    // 1) in_proj: [1024,8192] = hidden @ in_w^T; split x_pre | z
    {
        const int tilesN = (2 * DINNER) / 64;           // 128
        const int waves  = (LSEQ / 32) * tilesN;        // 4096
        gemm_wmma_f32<DMODEL, 4, 2, false, false, true>
            <<<waves / 8, BLK, 0, stream>>>(
                hidden, DMODEL, in_w,
                xpre, DINNER, ws_z, DINNER, nullptr, tilesN);
    }

    // 2) depthwise conv + silu -> x
    conv_silu_kernel<<<(LSEQ * DINNER) / BLK, BLK, 0, stream>>>(
        xpre, conv_w, conv_b, ws_x);

    // 3) x_proj: [1024,160] = x @ x_proj_w^T   (JSUB=2: 5 strips of 32 cols)
    {
        const int tilesN = XPDIM / 32;                  // 5
        const int waves  = (LSEQ / 32) * tilesN;        // 160
        gemm_wmma_f32<DINNER, 2, 2, false, false, false>
            <<<waves * 32 / BLK, BLK, 0, stream>>>(
                ws_x, DINNER, xproj_w,
                ws_xp, XPDIM, nullptr, 0, nullptr, tilesN);
    }

    // 4) dt_proj: [1024,4096] = softplus(xp[:, :128] @ dt_w^T + dt_b)
    {
        const int tilesN = DINNER / 64;                 // 64
        const int waves  = (LSEQ / 32) * tilesN;        // 2048
        gemm_wmma_f32<DTRANK, 4, 2, true, true, false>
            <<<waves / 8, BLK, 0, stream>>>(
                ws_xp, XPDIM, dt_w,
                ws_dt, DINNER, nullptr, 0, dt_b, tilesN);
    }

    // 5) selective scan + gating -> yg
    scan_kernel<<<(DINNER * DSTATE) / BLK, BLK, 0, stream>>>(
        ws_dt, ws_xp, ws_x, ws_z, A_log, Dp, ws_yg);

    // 6) out_proj: [1024,2048] = yg @ out_w^T
    {
        const int tilesN = DMODEL / 64;                 // 32
        const int waves  = (LSEQ / 32) * tilesN;        // 1024
        gemm_wmma_f32<DINNER, 4, 2, false, false, false>
            <<<waves / 8, BLK, 0, stream>>>(
                ws_yg, DINNER, out_w,
                out, DMODEL, nullptr, 0, nullptr, tilesN);
    }
}